// StructuredTransform_67018669687391
// MI455X (gfx1250) — compile-verified
//
#include <hip/hip_runtime.h>
#include <math.h>

typedef __attribute__((ext_vector_type(2))) float v2f;
typedef __attribute__((ext_vector_type(8))) float v8f;

#define N_RES    8192
#define N_IN     4096
#define N_PAD    16384
#define SIDE     128     // N_PAD = 128 * 128, transform = H128 * Xm * H128
#define LDSS     132     // row stride in LDS floats: 132 mod 64 banks = 4 -> conflict-free strided reads
#define BLOCK    512     // 16 waves of 32
#define N_LAYERS 3

// Hadamard entry H128[r][c] = (-1)^popcount(r & c), generated in registers (H input never read)
__device__ __forceinline__ float hpm(int r, int c) {
    return (__popc(r & c) & 1) ? -1.0f : 1.0f;
}

__global__ __launch_bounds__(BLOCK)
void StructuredTransform_fwht_wmma_kernel(const float* __restrict__ state,
                                          const float* __restrict__ inputs,
                                          const float* __restrict__ D,
                                          const float* __restrict__ bias,
                                          float* __restrict__ out) {
    __shared__ float Xs[SIDE * LDSS];   // 128 x 132 floats = 66 KB (of 320 KB/WGP)

    const int tid = threadIdx.x;

    // ---- load: X = [0.9*state | 0.4*inputs | 0-pad], reshaped row-major 128x128 ----
    for (int i = tid; i < N_PAD; i += BLOCK) {
        float v = 0.0f;
        if (i < N_RES)              v = 0.9f * state[i];
        else if (i < N_RES + N_IN)  v = 0.4f * inputs[i - N_RES];
        Xs[(i >> 7) * LDSS + (i & 127)] = v;
    }
    __syncthreads();

    const int wave = tid >> 5;
    const int lane = tid & 31;
    const int half = lane >> 4;      // 0: lanes 0-15, 1: lanes 16-31
    const int m    = lane & 15;
    const int tr   = wave >> 1;          // tile row 0..7 (rows tr*16 .. tr*16+15)
    const int tc0  = (wave & 1) * 4;     // this wave covers tile cols tc0..tc0+3

    for (int layer = 0; layer < N_LAYERS; ++layer) {
        // ---- elementwise Rademacher sign flip: X *= D[layer] ----
        const float* Dl = D + layer * N_PAD;
        for (int i = tid; i < N_PAD; i += BLOCK) {
            Xs[(i >> 7) * LDSS + (i & 127)] *= Dl[i];
        }
        __syncthreads();

        // ---- GEMM 1: T = H128 * X   (A generated, B from LDS) ----
        {
            v8f acc[4] = {};
            for (int kb = 0; kb < 32; ++kb) {
                const int k0 = kb * 4 + 2 * half;
                v2f a;                       // A[m][k0], A[m][k0+1]
                a.x = hpm(tr * 16 + m, k0);
                a.y = hpm(tr * 16 + m, k0 + 1);
                #pragma unroll
                for (int j = 0; j < 4; ++j) {
                    const int tc = tc0 + j;
                    v2f b;                   // B[k0][n], B[k0+1][n], n = m
                    b.x = Xs[k0       * LDSS + tc * 16 + m];
                    b.y = Xs[(k0 + 1) * LDSS + tc * 16 + m];
                    acc[j] = __builtin_amdgcn_wmma_f32_16x16x4_f32(
                        false, a, false, b, (short)0, acc[j], false, false);
                }
            }
            __syncthreads();   // all waves done reading X before in-place overwrite
            #pragma unroll
            for (int j = 0; j < 4; ++j) {
                const int tc = tc0 + j;
                #pragma unroll
                for (int i = 0; i < 8; ++i)   // row = tr*16 + half*8 + i, col = tc*16 + m
                    Xs[(tr * 16 + half * 8 + i) * LDSS + tc * 16 + m] = acc[j][i];
            }
            __syncthreads();
        }

        // ---- GEMM 2: X = T * H128   (A from LDS, B generated) ----
        {
            v8f acc[4] = {};
            for (int kb = 0; kb < 32; ++kb) {
                const int k0 = kb * 4 + 2 * half;
                v2f a;                       // A[m][k0], A[m][k0+1] from LDS (stride-132 rows)
                a.x = Xs[(tr * 16 + m) * LDSS + k0];
                a.y = Xs[(tr * 16 + m) * LDSS + k0 + 1];
                #pragma unroll
                for (int j = 0; j < 4; ++j) {
                    const int tc = tc0 + j;
                    v2f b;                   // B[k0][n] = H128[k0][tc*16+m]
                    b.x = hpm(k0,     tc * 16 + m);
                    b.y = hpm(k0 + 1, tc * 16 + m);
                    acc[j] = __builtin_amdgcn_wmma_f32_16x16x4_f32(
                        false, a, false, b, (short)0, acc[j], false, false);
                }
            }
            __syncthreads();
            #pragma unroll
            for (int j = 0; j < 4; ++j) {
                const int tc = tc0 + j;
                #pragma unroll
                for (int i = 0; i < 8; ++i)
                    Xs[(tr * 16 + half * 8 + i) * LDSS + tc * 16 + m] = acc[j][i];
            }
            __syncthreads();
        }
    }

    // ---- epilogue: out = erf(X[:8192]/16384 + bias)  (LEAK == 1.0) ----
    for (int j = tid; j < N_RES; j += BLOCK) {
        float x = Xs[(j >> 7) * LDSS + (j & 127)] * (1.0f / 16384.0f) + bias[j];
        out[j] = erff(x);
    }
}

extern "C" void kernel_launch(void* const* d_in, const int* in_sizes, int n_in,
                              void* d_out, int out_size, void* d_ws, size_t ws_size,
                              hipStream_t stream) {
    (void)in_sizes; (void)n_in; (void)d_ws; (void)ws_size; (void)out_size;
    const float* state  = (const float*)d_in[0];   // [1, 8192]
    const float* inputs = (const float*)d_in[1];   // [1, 4096]
    const float* D      = (const float*)d_in[2];   // [3, 16384]
    // d_in[3] = H (16384x16384) -- intentionally unused: generated analytically
    const float* bias   = (const float*)d_in[4];   // [8192]

    StructuredTransform_fwht_wmma_kernel<<<1, BLOCK, 0, stream>>>(
        state, inputs, D, bias, (float*)d_out);
}